// WaveletMixing_38955353374742
// MI455X (gfx1250) — compile-verified
//
#include <hip/hip_runtime.h>

// ---------------------------------------------------------------------------
// WaveletMixing on MI455X (gfx1250): single-pass, LDS-resident db4 DWT/IDWT.
// All convolutions are executed as banded-Toeplitz matmuls on
// V_WMMA_F32_16X16X4_F32 (wave32). 302KB LDS per workgroup (<=320KB/WGP).
// HBM traffic ~= read x + write out (+ L2-resident x re-read) -> ~10us bound.
// ---------------------------------------------------------------------------

typedef float v2f __attribute__((ext_vector_type(2)));
typedef float v8f __attribute__((ext_vector_type(8)));

constexpr int B_ = 8, N_ = 4096, D_ = 768;
constexpr int CH  = 8;   // channels per workgroup
constexpr int CHS = 9;   // LDS stride per position (padded: conflict-free)

constexpr int L0 = 4096, L1 = 2051, L2 = 1029, L3 = 518;

// LDS buffer offsets in "positions" (1 position = CHS floats)
constexpr int OFF_SX0 = 0;     // alloc 4168 : ext signal lvl1 (values at +6, valid raw 0..4108)
constexpr int OFF_A1  = 4168;  // alloc 2144 : cA1 ext (values at +6, valid raw 0..2063)
constexpr int OFF_D1  = 6312;  // alloc 2080 : cD1 plain
constexpr int OFF_A2  = 0;     // alloc 1120 : reuse SX0 region (values at +6)
constexpr int OFF_D2  = 1120;  // alloc 1056
constexpr int OFF_A3  = 2176;  // alloc 544  : plain
constexpr int OFF_D3  = 2720;  // alloc 544  : plain
constexpr int OFF_R3  = 0;     // alloc 1056 : reuse A2 region (plain)
constexpr int OFF_R2  = 4168;  // alloc 2144 : reuse A1 region (plain)
constexpr int LDS_POS = 8392;  // * CHS * 4B = 302,112 bytes

__constant__ float c_DLO[8] = {
    -0.010597401784997278f,  0.032883011666982945f,  0.030841381835986965f,
    -0.18703481171888114f,  -0.02798376941698385f,   0.6308807679295904f,
     0.7148465705525415f,    0.23037781330885523f };

static __device__ inline v8f wmma4(v2f a, v2f b, v8f c) {
  // (neg_a, A, neg_b, B, c_mod, C, reuse_a, reuse_b)
  return __builtin_amdgcn_wmma_f32_16x16x4_f32(false, a, false, b, (short)0, c,
                                               false, false);
}

// ---- small LDS utility phases (all 256 threads) ---------------------------
static __device__ inline void lds_fill_ext(float* raw, int L) {
  // raw: position-indexed (stride CHS); values at raw[(6+p)*CHS+ch]
  for (int e = threadIdx.x; e < 6 * CH; e += blockDim.x) {     // left: ext[j]=v[5-j]
    int j = e >> 3, ch = e & 7;
    raw[j * CHS + ch] = raw[(11 - j) * CHS + ch];
  }
  for (int e = threadIdx.x; e < 7 * CH; e += blockDim.x) {     // right: ext[6+L+r]=v[L-1-r]
    int r = e >> 3, ch = e & 7;
    raw[(6 + L + r) * CHS + ch] = raw[(5 + L - r) * CHS + ch];
  }
}
static __device__ inline void lds_zero(float* base, int p0, int p1) {
  for (int e = p0 * CH + threadIdx.x; e < p1 * CH; e += blockDim.x) {
    int p = e >> 3, ch = e & 7;
    base[p * CHS + ch] = 0.f;
  }
}

// ---- analysis: 32 output positions x 8 channels, both bands, per chain ----
template <bool FULL>
static __device__ inline void analysis_chain(const float* sx, float* oA, float* oD,
    int p0, int Lout, const v2f* alo, const v2f* ahi, float wA, float wD, int lane) {
  const int nn = lane & 15, t = nn >> 3, ch = nn & 7;
  const int kh = (lane >> 4) << 1;
  const int base = 2 * p0 + 32 * t + kh;
  v8f clo = {0,0,0,0,0,0,0,0}, chi = {0,0,0,0,0,0,0,0};
#pragma unroll
  for (int q = 0; q < 10; ++q) {
    const int idx = base + 4 * q;
    v2f bv; bv.x = sx[idx * CHS + ch]; bv.y = sx[(idx + 1) * CHS + ch];
    clo = wmma4(alo[q], bv, clo);
    chi = wmma4(ahi[q], bv, chi);
  }
  const int pb = p0 + 16 * t + ((lane >> 4) << 3);
#pragma unroll
  for (int r = 0; r < 8; ++r) {
    const int p = pb + r;
    if (FULL || p < Lout) {
      oA[p * CHS + ch] = clo[r] * wA;
      oD[p * CHS + ch] = chi[r] * wD;
    }
  }
}

// ---- synthesis accumulate: 32 outputs x 8 channels per chain --------------
static __device__ inline v8f synth_acc(const float* ca, const float* cd,
                                       int n0, const v2f* as, int lane) {
  const int nn = lane & 15, t = nn >> 3, ch = nn & 7;
  const int kh = (lane >> 4) << 1;
  const int hb = (n0 + 16 * t + kh) >> 1;   // cc base is even -> x from cA, y from cD
  v8f acc = {0,0,0,0,0,0,0,0};
#pragma unroll
  for (int q = 0; q < 6; ++q) {
    const int h = hb + 2 * q;
    v2f bv; bv.x = ca[h * CHS + ch]; bv.y = cd[h * CHS + ch];
    acc = wmma4(as[q], bv, acc);
  }
  return acc;
}
template <bool FULL>
static __device__ inline void synthesis_chain(const float* ca, const float* cd,
    float* o, int n0, int Lout, const v2f* as, int lane) {
  v8f acc = synth_acc(ca, cd, n0, as, lane);
  const int ch = lane & 7;
  const int pb = n0 + 16 * (((lane & 15) >> 3)) + ((lane >> 4) << 3);
#pragma unroll
  for (int r = 0; r < 8; ++r) {
    const int p = pb + r;
    if (FULL || p < Lout) o[p * CHS + ch] = acc[r];
  }
}
static __device__ inline void synthesis_global(const float* ca, const float* cd,
    const float* __restrict__ x, float* __restrict__ out, size_t xbase,
    int n0, const v2f* as, int lane) {
  v8f acc = synth_acc(ca, cd, n0, as, lane);
  const int ch = lane & 7;
  const int pb = n0 + 16 * (((lane & 15) >> 3)) + ((lane >> 4) << 3);
#pragma unroll
  for (int r = 0; r < 8; ++r) {
    const size_t g = xbase + (size_t)(pb + r) * D_ + ch;
    out[g] = x[g] + acc[r];   // residual fused, rec1 never stored
  }
}

__global__ __launch_bounds__(256)
void WaveletMixing_38955353374742_kernel(const float* __restrict__ x,
                                         const float* __restrict__ w_approx,
                                         const float* __restrict__ w_details,
                                         float* __restrict__ out) {
  __shared__ float smem[LDS_POS * CHS];
  const int tid  = threadIdx.x;
  const int lane = tid & 31;
  const int wave = tid >> 5;
  const int b     = blockIdx.x / (D_ / CH);
  const int dbase = (blockIdx.x % (D_ / CH)) * CH;
  const int ch    = tid & 7;

  // analysis kernels (cross-correlation taps = filter reversed) + synthesis taps
  float kLO[8], kHI[8], DHI8[8];
#pragma unroll
  for (int i = 0; i < 8; ++i) {
    kLO[i]  = c_DLO[7 - i];                              // DEC_LO[::-1]
    kHI[i]  = ((i & 1) ? -1.f : 1.f) * c_DLO[i];         // DEC_HI[::-1]
    DHI8[i] = ((i & 1) ?  1.f : -1.f) * c_DLO[7 - i];    // DEC_HI
  }

  // per-lane constant WMMA A-matrices (Toeplitz filter chunks, K=4 each)
  const int m  = lane & 15;
  const int kh = (lane >> 4) << 1;
  v2f ALO[10], AHI[10], AS[6];
#pragma unroll
  for (int q = 0; q < 10; ++q) {           // analysis: A[m][k] = kf[k-2m]
    const int k0 = 4 * q + kh;
    const int i0 = k0 - 2 * m, i1 = i0 + 1;
    ALO[q].x = (i0 >= 0 && i0 < 8) ? kLO[i0] : 0.f;
    ALO[q].y = (i1 >= 0 && i1 < 8) ? kLO[i1] : 0.f;
    AHI[q].x = (i0 >= 0 && i0 < 8) ? kHI[i0] : 0.f;
    AHI[q].y = (i1 >= 0 && i1 < 8) ? kHI[i1] : 0.f;
  }
#pragma unroll
  for (int q = 0; q < 6; ++q) {            // synthesis: A[m][k]=(k even?DLO:DHI)[k-m+1]
    const int k0 = 4 * q + kh;             // even
    const int i0 = k0 - m + 1;             // -> DEC_LO
    const int i1 = k0 + 2 - m;             // (k0+1)-m+1 -> DEC_HI
    AS[q].x = (i0 >= 0 && i0 < 8) ? c_DLO[i0] : 0.f;
    AS[q].y = (i1 >= 0 && i1 < 8) ? DHI8[i1]  : 0.f;
  }

  const float wapp = w_approx[dbase + ch];
  const float wd0  = w_details[0 * D_ + dbase + ch];
  const float wd1  = w_details[1 * D_ + dbase + ch];
  const float wd2  = w_details[2 * D_ + dbase + ch];

  const size_t xbase = ((size_t)b * N_) * D_ + dbase;

  // -- load x tile into extended-signal buffer ------------------------------
  {
    float* sx = smem + OFF_SX0 * CHS;
    for (int n = tid >> 3; n < N_; n += 32)
      sx[(6 + n) * CHS + ch] = x[xbase + (size_t)n * D_ + ch];
  }
  __syncthreads();
  lds_fill_ext(smem + OFF_SX0 * CHS, L0);
  lds_zero(smem + OFF_SX0 * CHS, 6 + L0 + 7, 4168);
  __syncthreads();

  // -- level-1 analysis: SX0 -> cA1(+6), cD1*wd0 ----------------------------
  {
    const float* sx = smem + OFF_SX0 * CHS;
    float* oA = smem + (OFF_A1 + 6) * CHS;
    float* oD = smem + OFF_D1 * CHS;
    const int nfull = L1 >> 5, ntot = (L1 + 31) >> 5;
    for (int c = wave; c < nfull; c += 8)
      analysis_chain<true >(sx, oA, oD, c * 32, L1, ALO, AHI, 1.f, wd0, lane);
    for (int c = nfull + wave; c < ntot; c += 8)
      analysis_chain<false>(sx, oA, oD, c * 32, L1, ALO, AHI, 1.f, wd0, lane);
  }
  __syncthreads();
  lds_fill_ext(smem + OFF_A1 * CHS, L1);
  lds_zero(smem + OFF_A1 * CHS, 6 + L1 + 7, 2144);
  lds_zero(smem + OFF_D1 * CHS, L1, 2080);
  __syncthreads();

  // -- level-2 analysis: cA1 -> cA2(+6), cD2*wd1 ----------------------------
  {
    const float* sx = smem + OFF_A1 * CHS;
    float* oA = smem + (OFF_A2 + 6) * CHS;
    float* oD = smem + OFF_D2 * CHS;
    const int nfull = L2 >> 5, ntot = (L2 + 31) >> 5;
    for (int c = wave; c < nfull; c += 8)
      analysis_chain<true >(sx, oA, oD, c * 32, L2, ALO, AHI, 1.f, wd1, lane);
    for (int c = nfull + wave; c < ntot; c += 8)
      analysis_chain<false>(sx, oA, oD, c * 32, L2, ALO, AHI, 1.f, wd1, lane);
  }
  __syncthreads();
  lds_fill_ext(smem + OFF_A2 * CHS, L2);
  lds_zero(smem + OFF_A2 * CHS, 6 + L2 + 7, 1120);
  lds_zero(smem + OFF_D2 * CHS, L2, 1056);
  __syncthreads();

  // -- level-3 analysis: cA2 -> cA3*wapp, cD3*wd2 (plain layout) ------------
  {
    const float* sx = smem + OFF_A2 * CHS;
    float* oA = smem + OFF_A3 * CHS;
    float* oD = smem + OFF_D3 * CHS;
    const int nfull = L3 >> 5, ntot = (L3 + 31) >> 5;
    for (int c = wave; c < nfull; c += 8)
      analysis_chain<true >(sx, oA, oD, c * 32, L3, ALO, AHI, wapp, wd2, lane);
    for (int c = nfull + wave; c < ntot; c += 8)
      analysis_chain<false>(sx, oA, oD, c * 32, L3, ALO, AHI, wapp, wd2, lane);
  }
  __syncthreads();
  lds_zero(smem + OFF_A3 * CHS, L3, 544);
  lds_zero(smem + OFF_D3 * CHS, L3, 544);
  __syncthreads();

  // -- synthesis level 3: (cA3, cD3) -> rec3 (len L2) -----------------------
  {
    const float* ca = smem + OFF_A3 * CHS;
    const float* cd = smem + OFF_D3 * CHS;
    float* o = smem + OFF_R3 * CHS;
    const int nfull = L2 >> 5, ntot = (L2 + 31) >> 5;
    for (int c = wave; c < nfull; c += 8)
      synthesis_chain<true >(ca, cd, o, c * 32, L2, AS, lane);
    for (int c = nfull + wave; c < ntot; c += 8)
      synthesis_chain<false>(ca, cd, o, c * 32, L2, AS, lane);
  }
  __syncthreads();
  lds_zero(smem + OFF_R3 * CHS, L2, 1056);
  __syncthreads();

  // -- synthesis level 2: (rec3, cD2) -> rec2 (len L1) ----------------------
  {
    const float* ca = smem + OFF_R3 * CHS;
    const float* cd = smem + OFF_D2 * CHS;
    float* o = smem + OFF_R2 * CHS;
    const int nfull = L1 >> 5, ntot = (L1 + 31) >> 5;
    for (int c = wave; c < nfull; c += 8)
      synthesis_chain<true >(ca, cd, o, c * 32, L1, AS, lane);
    for (int c = nfull + wave; c < ntot; c += 8)
      synthesis_chain<false>(ca, cd, o, c * 32, L1, AS, lane);
  }
  __syncthreads();
  lds_zero(smem + OFF_R2 * CHS, L1, L1 + 8);
  __syncthreads();

  // -- synthesis level 1: (rec2, cD1) -> out = x + rec1 (fused to HBM) ------
  for (int c = wave; c < (L0 >> 5); c += 8)   // 4096 = 128 full chains
    synthesis_global(smem + OFF_R2 * CHS, smem + OFF_D1 * CHS,
                     x, out, xbase, c * 32, AS, lane);
}

extern "C" void kernel_launch(void* const* d_in, const int* in_sizes, int n_in,
                              void* d_out, int out_size, void* d_ws, size_t ws_size,
                              hipStream_t stream) {
  (void)in_sizes; (void)n_in; (void)d_ws; (void)ws_size; (void)out_size;
  const float* x  = (const float*)d_in[0];
  const float* wa = (const float*)d_in[1];
  const float* wd = (const float*)d_in[2];
  float* out = (float*)d_out;
  dim3 grid(B_ * (D_ / CH));   // 768 workgroups, one per (batch, 8-channel block)
  dim3 block(256);             // 8 waves of 32
  hipLaunchKernelGGL(WaveletMixing_38955353374742_kernel, grid, block, 0, stream,
                     x, wa, wd, out);
}